// activation_sub_up_47167330845364
// MI455X (gfx1250) — compile-verified
//
#include <hip/hip_runtime.h>
#include <math.h>

#ifndef __has_builtin
#define __has_builtin(x) 0
#endif

// Native clang vector type: accepted by __builtin_nontemporal_load/store and
// lowers to single global_load_b128 / global_store_b128.
typedef __attribute__((ext_vector_type(4))) float v4f;

__device__ __forceinline__ float htanh(float x) {
#if __has_builtin(__builtin_amdgcn_tanhf)
  return __builtin_amdgcn_tanhf(x);          // v_tanh_f32 (gfx1250 TRANS op)
#elif __has_builtin(__builtin_amdgcn_tanh_f32)
  return __builtin_amdgcn_tanh_f32(x);
#else
  return tanhf(x);                           // fallback: ocml tanh
#endif
}

// Layout constants fixed by the reference: DIM = 2048 floats per half-row.
// Half-row = 512 float4s, full row = 1024 float4s.
#define HALF4_SHIFT 9    // 512 float4 per half-row
#define ROW4_SHIFT  10   // 1024 float4 per row

__global__ __launch_bounds__(256) void pq_tanh_update_kernel(
    const v4f* __restrict__ pq4,   // (rows, 1024) float4
    const v4f* __restrict__ a4,    // (512,) float4  -- 8 KB, stays in cache
    v4f* __restrict__ out4,        // (rows, 1024) float4
    int total)                     // rows * 512
{
  int t = blockIdx.x * blockDim.x + threadIdx.x;
  if (t >= total) return;

  int row  = t >> HALF4_SHIFT;
  int col  = t & ((1 << HALF4_SHIFT) - 1);
  int base = row << ROW4_SHIFT;

  // Streaming reads: each element touched exactly once -> non-temporal.
  v4f p = __builtin_nontemporal_load(&pq4[base + col]);
  v4f q = __builtin_nontemporal_load(&pq4[base + (1 << HALF4_SHIFT) + col]);
  // 'a' is reused by every row -> regular (temporal) load, hits WGP$/L2.
  v4f av = a4[col];

  v4f r;
  r.x = fmaf(av.x, htanh(q.x), p.x);
  r.y = fmaf(av.y, htanh(q.y), p.y);
  r.z = fmaf(av.z, htanh(q.z), p.z);
  r.w = fmaf(av.w, htanh(q.w), p.w);

  // p_new into first half, q passthrough into second half. NT stores: the
  // output is write-once, never re-read by this dispatch.
  __builtin_nontemporal_store(r, &out4[base + col]);
  __builtin_nontemporal_store(q, &out4[base + (1 << HALF4_SHIFT) + col]);
}

extern "C" void kernel_launch(void* const* d_in, const int* in_sizes, int n_in,
                              void* d_out, int out_size, void* d_ws, size_t ws_size,
                              hipStream_t stream) {
  const float* pq = (const float*)d_in[0];   // (16384, 4096) f32
  const float* a  = (const float*)d_in[1];   // (2048,) f32

  const int dim  = in_sizes[1];              // 2048 (fixed by reference)
  const int rows = in_sizes[0] / (2 * dim);  // 16384
  const int total = rows * (dim / 4);        // one thread per p-half float4

  const int block = 256;                     // 8 wave32 waves per workgroup
  const int grid  = (total + block - 1) / block;

  pq_tanh_update_kernel<<<grid, block, 0, stream>>>(
      (const v4f*)pq, (const v4f*)a, (v4f*)d_out, total);
}